// StdPool2D_89507118448970
// MI455X (gfx1250) — compile-verified
//
#include <hip/hip_runtime.h>

// StdPool2D: per-window std over 7x7 patches, stride 3, NCHW (8,64,256,256) f32.
// Memory-bound (142 MB traffic vs ~35 MFLOP): HBM floor ~6us at 23.3 TB/s.
// gfx1250 path: async global->LDS staging (no VGPR round trip, ASYNCcnt-tracked),
// separable (sum, sumsq) window reduction entirely in LDS, coalesced stores.

#define KSZ   7
#define STR   3
#define W_IN  256
#define H_IN  256
#define W_OUT 84
#define H_OUT 84
#define R_TILE 12                                   // output rows per workgroup
#define TILES_PER_PLANE (H_OUT / R_TILE)            // 7 (compile-time: no runtime div)
#define ROWS_IN_TILE (STR * R_TILE + (KSZ - STR))   // 40 input rows
#define IN_FLOATS (ROWS_IN_TILE * W_IN)             // 10240 floats = 40 KB
#define NTHREADS 256                                // 8 wave32s

__global__ void __launch_bounds__(NTHREADS)
stdpool2d_kernel(const float* __restrict__ x, float* __restrict__ out) {
    // LDS: 40 KB input stage + 26.25 KB separable (sum, sumsq) buffer = ~67 KB
    // -> ~4 workgroups (32 wave32s) co-resident per 320 KB WGP.
    __shared__ __align__(16) float  lds_in[IN_FLOATS];
    __shared__ __align__(16) float2 lds_h[ROWS_IN_TILE * W_OUT];

    const int tid   = threadIdx.x;
    const int tile  = blockIdx.x % TILES_PER_PLANE;   // const divisor -> mulhi
    const int plane = blockIdx.x / TILES_PER_PLANE;

    const int r_out0 = tile * R_TILE;
    const int r_in0  = r_out0 * STR;                  // first input row of tile

    const float* src = x + (size_t)plane * (H_IN * W_IN) + (size_t)r_in0 * W_IN;

    // ---- Phase 0: async-copy 40 contiguous rows (40 KB) global -> LDS ------
    // Per-lane b128 async loads: dsaddr VGPR = LDS byte offset, addr = VGPR pair.
    #pragma unroll
    for (int i = 0; i < IN_FLOATS / (NTHREADS * 4); ++i) {   // 10 rounds
        const int f = (i * NTHREADS + tid) * 4;              // 16B-aligned float idx
        unsigned lds_addr =
            (unsigned)(unsigned long long)(const void*)&lds_in[f];
        unsigned long long gaddr = (unsigned long long)(const void*)(src + f);
        asm volatile("global_load_async_to_lds_b128 %0, %1, off"
                     :: "v"(lds_addr), "v"(gaddr)
                     : "memory");
    }
    asm volatile("s_wait_asynccnt 0" ::: "memory");   // this wave's copies done
    __syncthreads();                                  // all waves' copies visible

    // ---- Phase A: horizontal 7-wide sums of (x, x^2) per input row ---------
    // 40 rows x 84 cols = 3360 tasks; carry (row,col) incrementally, no divides
    // in the loop body. Stride-3 LDS reads are bank-conflict-free (3 ⟂ 64).
    {
        int row = tid / W_OUT;            // one const-divisor div at entry
        int col = tid - row * W_OUT;
        for (int t = tid; t < ROWS_IN_TILE * W_OUT; t += NTHREADS) {
            const float* p = &lds_in[row * W_IN + col * STR];
            float s = 0.0f, ss = 0.0f;
            #pragma unroll
            for (int k = 0; k < KSZ; ++k) {
                const float v = p[k];
                s += v;
                ss = fmaf(v, v, ss);
            }
            lds_h[t] = make_float2(s, ss);
            // advance by NTHREADS = 3*84 + 4 -> row += 3, col += 4 (w/ carry)
            row += NTHREADS / W_OUT;
            col += NTHREADS % W_OUT;
            if (col >= W_OUT) { col -= W_OUT; ++row; }
        }
    }
    __syncthreads();

    // ---- Phase B: vertical 7-row reduction + finalize ----------------------
    float* dst = out + (size_t)plane * (H_OUT * W_OUT) + (size_t)r_out0 * W_OUT;
    const float inv_n = 1.0f / (float)(KSZ * KSZ);
    {
        int orow = tid / W_OUT;
        int col  = tid - orow * W_OUT;
        for (int u = tid; u < R_TILE * W_OUT; u += NTHREADS) {
            const float2* hp = &lds_h[(orow * STR) * W_OUT + col];
            float s = 0.0f, ss = 0.0f;
            #pragma unroll
            for (int k = 0; k < KSZ; ++k) {
                const float2 h = hp[k * W_OUT];
                s  += h.x;
                ss += h.y;
            }
            const float mean = s * inv_n;
            float var = fmaf(-mean, mean, ss * inv_n);  // E[x^2] - E[x]^2
            var = fmaxf(var, 0.0f);
            dst[u] = sqrtf(var);                        // lanes -> cols: coalesced
            orow += NTHREADS / W_OUT;
            col  += NTHREADS % W_OUT;
            if (col >= W_OUT) { col -= W_OUT; ++orow; }
        }
    }
}

extern "C" void kernel_launch(void* const* d_in, const int* in_sizes, int n_in,
                              void* d_out, int out_size, void* d_ws, size_t ws_size,
                              hipStream_t stream) {
    (void)n_in; (void)out_size; (void)d_ws; (void)ws_size;
    const float* x = (const float*)d_in[0];
    float* out = (float*)d_out;

    const int planes  = in_sizes[0] / (H_IN * W_IN);      // B*C = 512
    const int nblocks = planes * TILES_PER_PLANE;         // 3584 workgroups

    stdpool2d_kernel<<<nblocks, NTHREADS, 0, stream>>>(x, out);
}